// ANR_ARL_12446815224135
// MI455X (gfx1250) — compile-verified
//
#include <hip/hip_runtime.h>

typedef float v2f __attribute__((ext_vector_type(2)));
typedef float v8f __attribute__((ext_vector_type(8)));

#define B_ 64
#define S_ 1024
#define H_ 128
#define A_ 8
#define W_ 5
#define AW 40      // A_*W_
#define NT 48      // t columns padded to 3 * 16

// ---------------------------------------------------------------------------
// ws layout (floats):
//   u : [0,      6144)                48x128, padded aspect/window vectors
//   c : [8192,   8192+65536)          per-(a,b) weighted docIn sums (A,B,H)
//   t : [73728,  73728+65536*48)      logits GEMM output (B*S, 48)
// ---------------------------------------------------------------------------

// u[n=a*5+w][h] = sum_f aspProj[a,h,f] * E[a,f,w];  E[a,f,w] = aspE[a, f*5+w]
__global__ void k1_u(const float* __restrict__ aspProj,
                     const float* __restrict__ aspE,
                     float* __restrict__ u) {
  int n = blockIdx.x;       // 0..47 (40..47 = zero pad)
  int h = threadIdx.x;      // 0..127
  float acc = 0.f;
  if (n < AW) {
    int a = n / W_, w = n % W_;
    const float* P = aspProj + (size_t)a * H_ * H_ + (size_t)h * H_;
    const float* E = aspE + (size_t)a * (W_ * H_);
#pragma unroll 8
    for (int f = 0; f < H_; ++f) acc += P[f] * E[f * W_ + w];
  }
  u[(size_t)n * H_ + h] = acc;
}

__global__ void k0_zero(float* __restrict__ c) {
  c[(size_t)blockIdx.x * 256 + threadIdx.x] = 0.f;
}

// t[(b*S+s), n] = docIn[b,s,:] . u[n,:]   -- (65536 x 128) x (128 x 48) GEMM
// One wave per 16-row M tile; A fragment reused across the 3 N tiles.
__global__ void k2_logits(const float* __restrict__ docIn,
                          const float* __restrict__ u,
                          float* __restrict__ t) {
  int lane = threadIdx.x & 31;
  int wave = threadIdx.x >> 5;
  int mtile = blockIdx.x * 8 + wave;        // 0..4095
  int half = lane >> 4;                     // K-half select per ISA layout
  int col  = lane & 15;
  int row  = mtile * 16 + (lane & 15);
  const float* arow = docIn + (size_t)row * H_;
  v8f acc0 = {}, acc1 = {}, acc2 = {};
#pragma unroll
  for (int kc = 0; kc < H_; kc += 4) {
    int k = kc + 2 * half;
    v2f av = *(const v2f*)(arow + k);
    v2f b0 = *(const v2f*)(u + (size_t)(0 * 16 + col) * H_ + k);
    v2f b1 = *(const v2f*)(u + (size_t)(1 * 16 + col) * H_ + k);
    v2f b2 = *(const v2f*)(u + (size_t)(2 * 16 + col) * H_ + k);
    acc0 = __builtin_amdgcn_wmma_f32_16x16x4_f32(false, av, false, b0,
                                                 (short)0, acc0, false, false);
    acc1 = __builtin_amdgcn_wmma_f32_16x16x4_f32(false, av, false, b1,
                                                 (short)0, acc1, false, false);
    acc2 = __builtin_amdgcn_wmma_f32_16x16x4_f32(false, av, false, b2,
                                                 (short)0, acc2, false, false);
  }
  int mbase = mtile * 16;
#pragma unroll
  for (int r = 0; r < 8; ++r) {
    int m = mbase + r + 8 * half;
    float* trow = t + (size_t)m * NT;
    trow[0 * 16 + col] = acc0[r];
    trow[1 * 16 + col] = acc1[r];
    trow[2 * 16 + col] = acc2[r];
  }
}

// 5-tap shifted sum over t, softmax over s, write attn in (B,A,S) order.
__global__ void k3_softmax(const float* __restrict__ t,
                           float* __restrict__ attn) {
  int b = blockIdx.x >> 3;
  int a = blockIdx.x & 7;
  int tid = threadIdx.x;                    // 256 threads, 4 s-values each
  __shared__ float red[256];

  float lg[4];
  float lmax = -3.402823466e38f;
#pragma unroll
  for (int i = 0; i < 4; ++i) {
    int s = tid + i * 256;
    float v = 0.f;
#pragma unroll
    for (int w = 0; w < W_; ++w) {
      int sp = s + w - 2;
      if (sp >= 0 && sp < S_)
        v += t[(size_t)(b * S_ + sp) * NT + a * W_ + w];
    }
    lg[i] = v;
    lmax = fmaxf(lmax, v);
  }
  red[tid] = lmax;
  __syncthreads();
  for (int off = 128; off > 0; off >>= 1) {
    if (tid < off) red[tid] = fmaxf(red[tid], red[tid + off]);
    __syncthreads();
  }
  float gmax = red[0];
  __syncthreads();

  float lsum = 0.f;
#pragma unroll
  for (int i = 0; i < 4; ++i) {
    lg[i] = __expf(lg[i] - gmax);
    lsum += lg[i];
  }
  red[tid] = lsum;
  __syncthreads();
  for (int off = 128; off > 0; off >>= 1) {
    if (tid < off) red[tid] += red[tid + off];
    __syncthreads();
  }
  float inv = 1.f / red[0];

  float* o = attn + (size_t)b * A_ * S_ + (size_t)a * S_;
#pragma unroll
  for (int i = 0; i < 4; ++i) o[tid + i * 256] = lg[i] * inv;
}

// c[a,b,h] += sum_s attn[b,a,s] * docIn[b,s,h]  (docIn read once; 8 aspects/load)
__global__ void k4_c(const float* __restrict__ docIn,
                     const float* __restrict__ attn,
                     float* __restrict__ c) {
  int b = blockIdx.x;
  int chunk = blockIdx.y;                   // 0..7, 128 s-values each
  int h = threadIdx.x;                      // 0..127
  float acc[A_];
#pragma unroll
  for (int a = 0; a < A_; ++a) acc[a] = 0.f;
  int s0 = chunk * 128;
  const float* dbase = docIn + ((size_t)b * S_ + s0) * H_ + h;
  const float* abase = attn + (size_t)b * A_ * S_ + s0;
  for (int si = 0; si < 128; ++si) {
    float d = dbase[(size_t)si * H_];
#pragma unroll
    for (int a = 0; a < A_; ++a) acc[a] += abase[(size_t)a * S_ + si] * d;
  }
#pragma unroll
  for (int a = 0; a < A_; ++a)
    atomicAdd(&c[(size_t)a * B_ * H_ + (size_t)b * H_ + h], acc[a]);
}

// rep[b,a,f] = c[a,b,:] @ aspProj[a,:,f] -- 8 GEMMs of 64x128x128 via WMMA f32
__global__ void k5_rep(const float* __restrict__ c,
                       const float* __restrict__ aspProj,
                       float* __restrict__ rep) {
  int lane = threadIdx.x & 31;
  int wave = threadIdx.x >> 5;
  int wg = blockIdx.x * 8 + wave;           // 0..255
  int a = wg >> 5;
  int rem = wg & 31;
  int mtile = rem >> 3;                     // 0..3  (b tiles)
  int ntile = rem & 7;                      // 0..7  (f tiles)
  int half = lane >> 4;
  int col  = lane & 15;
  int row  = mtile * 16 + (lane & 15);
  const float* arow = c + (size_t)a * B_ * H_ + (size_t)row * H_;
  const float* Bm = aspProj + (size_t)a * H_ * H_;
  int n = ntile * 16 + col;
  v8f acc = {};
#pragma unroll
  for (int kc = 0; kc < H_; kc += 4) {
    int k = kc + 2 * half;
    v2f av = *(const v2f*)(arow + k);
    v2f bv;
    bv.x = Bm[(size_t)k * H_ + n];
    bv.y = Bm[(size_t)(k + 1) * H_ + n];
    acc = __builtin_amdgcn_wmma_f32_16x16x4_f32(false, av, false, bv,
                                                (short)0, acc, false, false);
  }
#pragma unroll
  for (int r = 0; r < 8; ++r) {
    int bb = mtile * 16 + r + 8 * half;
    rep[((size_t)bb * A_ + a) * H_ + n] = acc[r];
  }
}

extern "C" void kernel_launch(void* const* d_in, const int* in_sizes, int n_in,
                              void* d_out, int out_size, void* d_ws, size_t ws_size,
                              hipStream_t stream) {
  (void)in_sizes; (void)n_in; (void)out_size; (void)ws_size;
  const float* docIn   = (const float*)d_in[0];   // (B,S,H)
  const float* aspE    = (const float*)d_in[1];   // (A, W*H)
  const float* aspProj = (const float*)d_in[2];   // (A, H, H)

  float* attn = (float*)d_out;                    // (B,A,S)
  float* rep  = attn + (size_t)B_ * A_ * S_;      // (B,A,H)

  float* u = (float*)d_ws;                        // 6144 floats (+pad)
  float* cws = u + 8192;                          // 65536 floats
  float* t = cws + 65536;                         // 65536*48 floats

  k1_u<<<48, 128, 0, stream>>>(aspProj, aspE, u);
  k0_zero<<<256, 256, 0, stream>>>(cws);
  k2_logits<<<512, 256, 0, stream>>>(docIn, u, t);
  k3_softmax<<<512, 256, 0, stream>>>(t, attn);
  k4_c<<<dim3(64, 8), 128, 0, stream>>>(docIn, attn, cws);
  k5_rep<<<32, 256, 0, stream>>>(cws, aspProj, rep);
}